// MiniLM_85856396247916
// MI455X (gfx1250) — compile-verified
//
#include <hip/hip_runtime.h>

#define VOCAB  32000
#define HIDDEN 256
#define BATCH  16
#define TT     256
#define NROWS  (BATCH*TT)           // 4096 rows (b*T+t)
#define FC_ROWS  128                // rows per FC block (8 waves x 16)
#define FC_TILES 16                 // 16-wide N tiles per chunk
#define FC_CHUNK (FC_TILES*16)      // 256 vocab cols per chunk
#define NCHUNKS  (VOCAB/FC_CHUNK)   // 125
#define LDA 260                     // padded LDS row stride: 16B-aligned, banks 4l (conflict-free b64)
#define LDL 257                     // logits scratch stride

typedef __attribute__((ext_vector_type(2))) float v2f;
typedef __attribute__((ext_vector_type(4))) float v4f;
typedef __attribute__((ext_vector_type(8))) float v8f;

// Branchless tanh: no EXEC divergence on the sequential RNN path.
static __device__ __forceinline__ float fast_tanh(float x) {
  const float ax = fabsf(x);
  const float e  = __expf(-2.0f * ax);          // v_exp_f32 path, e in (0,1]
  const float y  = (1.0f - e) * __builtin_amdgcn_rcpf(1.0f + e);
  return copysignf(y, x);
}

// ---------------- Kernel 1: xproj[r][h] = Wih[tok[r]][h] + bih[h] + bhh[h] ----
__global__ void gather_kernel(const int* __restrict__ inputs,
                              const float* __restrict__ Wih,
                              const float* __restrict__ bih,
                              const float* __restrict__ bhh,
                              float* __restrict__ xproj) {
  const int r = blockIdx.x;        // r = b*T + t
  const int h = threadIdx.x;
  const int tok = inputs[r];
  xproj[r * HIDDEN + h] = Wih[tok * HIDDEN + h] + bih[h] + bhh[h];
}

// ---------------- Kernel 2: persistent single-WGP RNN scan --------------------
// Whh (256KB) in LDS (CDNA5: 320KB/WGP). 16 waves, wave w owns h-columns
// [16w,16w+16). Per step: 64x V_WMMA_F32_16X16X4_F32 over K=256, branchless
// tanh, double-buffered h. h buffer is placed FIRST in LDS so its addresses
// stay within the ds 2addr offset reach; xproj pipelined one step ahead.
__launch_bounds__(512, 1)
__global__ void rnn_kernel(const float* __restrict__ xproj,
                           const float* __restrict__ Whh,
                           float* __restrict__ hs) {
  extern __shared__ float smem[];
  float* hbuf = smem;                       // 2 * BATCH * LDA (double buffer)
  float* whh  = smem + 2 * BATCH * LDA;     // HIDDEN * LDA

  for (int i4 = threadIdx.x; i4 < HIDDEN * HIDDEN / 4; i4 += blockDim.x) {
    const int row = i4 >> 6;
    const int col = (i4 & 63) << 2;
    *(v4f*)&whh[row * LDA + col] = *(const v4f*)&Whh[row * HIDDEN + col];
  }
  for (int i = threadIdx.x; i < 2 * BATCH * LDA; i += blockDim.x) hbuf[i] = 0.f;
  __syncthreads();

  const int wave = threadIdx.x >> 5;
  const int lane = threadIdx.x & 31;
  const int half = lane >> 4;               // K sub-block selector
  const int l    = lane & 15;
  const int ncol = wave * 16 + l;           // output hidden column

  const float* wrow = &whh[ncol * LDA];     // Whh row base, hoisted

  // Per-lane xproj row pointers for the 8 batch rows this lane produces.
  const float* xpp[8];
#pragma unroll
  for (int i = 0; i < 8; ++i)
    xpp[i] = &xproj[((i + 8 * half) * TT) * HIDDEN + ncol];

  float xp[8];                              // xproj values for current step
#pragma unroll
  for (int i = 0; i < 8; ++i) xp[i] = xpp[i][0];     // t = 0

  int cur = 0;
  for (int t = 0; t < TT; ++t) {
    // Issue next step's xproj loads now; 64 WMMAs hide the latency.
    const int tn = (t + 1 < TT) ? (t + 1) : t;
    float xpn[8];
#pragma unroll
    for (int i = 0; i < 8; ++i) xpn[i] = xpp[i][(size_t)tn * HIDDEN];

    const float* hrow = &hbuf[cur * BATCH * LDA + l * LDA];
    v8f acc = {};
#pragma unroll 8
    for (int k = 0; k < HIDDEN; k += 4) {
      // A 16x4 f32: lane<16 -> {h[l][k],h[l][k+1]}, lane>=16 -> {h[l][k+2],h[l][k+3]}
      const v2f a = *(const v2f*)&hrow[k + 2 * half];
      // B 4x16 f32: B[k][n] = Whh[n][k] -> contiguous float2 from row ncol
      const v2f b = *(const v2f*)&wrow[k + 2 * half];
      acc = __builtin_amdgcn_wmma_f32_16x16x4_f32(false, a, false, b,
                                                  (short)0, acc, false, false);
    }
    float* hn = hbuf + (cur ^ 1) * BATCH * LDA;
#pragma unroll
    for (int i = 0; i < 8; ++i) {
      const int m = i + 8 * half;           // batch row (C/D layout)
      const float v = fast_tanh(acc[i] + xp[i]);
      hn[m * LDA + ncol] = v;
      hs[(m * TT + t) * HIDDEN + ncol] = v;
      xp[i] = xpn[i];
    }
    __syncthreads();
    cur ^= 1;
  }
}

// ---------------- Kernel 3: FC (hs @ Wfc^T + bfc) + partial log-softmax -------
// Block = 256 threads (8 waves): 128 rows x 256 vocab cols. Wfc tiles are
// double-buffered: next tile's global loads issue before the WMMA loop
// (latency hidden), ds_store to the alternate buffer after it.
__launch_bounds__(256, 1)
__global__ void fc_kernel(const float* __restrict__ hs,
                          const float* __restrict__ Wfc,
                          const float* __restrict__ bfc,
                          const int* __restrict__ targets,
                          float* __restrict__ pmax,
                          float* __restrict__ psum,
                          float* __restrict__ tlogit) {
  extern __shared__ float smem[];
  float* ldsA = smem;                        // FC_ROWS * LDA
  float* ldsB = ldsA + FC_ROWS * LDA;        // 2 * 16 * LDA (double buffer + stats scratch)
  float* ldsL = ldsB + 2 * 16 * LDA;         // FC_ROWS * LDL (chunk logits)

  const int tid    = threadIdx.x;
  const int r0     = blockIdx.x * FC_ROWS;
  const int v0base = blockIdx.y * FC_CHUNK;

  // Stage A: 128 rows of hs (float4, coalesced).
  for (int i4 = tid; i4 < FC_ROWS * HIDDEN / 4; i4 += blockDim.x) {
    const int row = i4 >> 6;
    const int col = (i4 & 63) << 2;
    *(v4f*)&ldsA[row * LDA + col] = *(const v4f*)&hs[(r0 + row) * HIDDEN + col];
  }
  const int wave = tid >> 5, lane = tid & 31, half = lane >> 4, l = lane & 15;
  const int arow = wave * 16 + l;            // this wave's A row for lane

  // This thread's 4 float4 slots of a 16x256 B tile.
  const int brow0 = tid >> 6;                // + j*4 per slot (j=0..3)
  const int bcol  = (tid & 63) << 2;

  v4f breg[4];
#pragma unroll
  for (int j = 0; j < 4; ++j)                // stage tile 0 into registers
    breg[j] = *(const v4f*)&Wfc[(v0base + brow0 + 4 * j) * HIDDEN + bcol];
#pragma unroll
  for (int j = 0; j < 4; ++j)
    *(v4f*)&ldsB[(brow0 + 4 * j) * LDA + bcol] = breg[j];
  __syncthreads();                           // covers A stage + B tile 0

  int bb = 0;
  for (int tile = 0; tile < FC_TILES; ++tile) {
    const int v0 = v0base + tile * 16;
    if (tile + 1 < FC_TILES) {               // issue next tile's loads early
#pragma unroll
      for (int j = 0; j < 4; ++j)
        breg[j] = *(const v4f*)&Wfc[(v0 + 16 + brow0 + 4 * j) * HIDDEN + bcol];
    }
    const float bias = bfc[v0 + l];          // issued before the K-loop
    const float* bbuf = &ldsB[bb * 16 * LDA + l * LDA];

    v8f acc = {};
#pragma unroll 8
    for (int k = 0; k < HIDDEN; k += 4) {
      const v2f a = *(const v2f*)&ldsA[arow * LDA + k + 2 * half];
      const v2f b = *(const v2f*)&bbuf[k + 2 * half];
      acc = __builtin_amdgcn_wmma_f32_16x16x4_f32(false, a, false, b,
                                                  (short)0, acc, false, false);
    }
#pragma unroll
    for (int i = 0; i < 8; ++i) {
      const int m = wave * 16 + i + 8 * half;          // local row 0..127
      ldsL[m * LDL + tile * 16 + l] = acc[i] + bias;
    }
    if (tile + 1 < FC_TILES) {               // commit next tile (loads now cheap)
      float* nbuf = &ldsB[(bb ^ 1) * 16 * LDA];
#pragma unroll
      for (int j = 0; j < 4; ++j)
        *(v4f*)&nbuf[(brow0 + 4 * j) * LDA + bcol] = breg[j];
    }
    __syncthreads();
    bb ^= 1;
  }

  // Partial softmax stats: 2 threads per row (128-col segments each).
  {
    const int row = tid & 127;
    const int seg = tid >> 7;                // 0 or 1
    const float* rowp = &ldsL[row * LDL + seg * 128];
    float m = -3.4e38f;
    for (int c = 0; c < 128; ++c) m = fmaxf(m, rowp[c]);
    float s = 0.f;
    for (int c = 0; c < 128; ++c) s += __expf(rowp[c] - m);
    float* stm = ldsB;                       // B buffers are dead now: reuse
    float* sts = ldsB + 256;
    stm[seg * 128 + row] = m;
    sts[seg * 128 + row] = s;
  }
  __syncthreads();
  if (tid < FC_ROWS) {
    const float* stm = ldsB;
    const float* sts = ldsB + 256;
    const float m0 = stm[tid], m1 = stm[128 + tid];
    const float s0 = sts[tid], s1 = sts[128 + tid];
    const float M = fmaxf(m0, m1);
    const float S = s0 * __expf(m0 - M) + s1 * __expf(m1 - M);
    const int r = r0 + tid;
    pmax[r * NCHUNKS + blockIdx.y] = M;
    psum[r * NCHUNKS + blockIdx.y] = S;
    const int tgt = targets[r];
    if (tgt >= v0base && tgt < v0base + FC_CHUNK)
      tlogit[r] = ldsL[tid * LDL + (tgt - v0base)];
  }
}

// ---------------- Kernel 4: combine partials -> mean NLL ----------------------
__launch_bounds__(512)
__global__ void loss_kernel(const float* __restrict__ pmax,
                            const float* __restrict__ psum,
                            const float* __restrict__ tlogit,
                            float* __restrict__ out) {
  __shared__ float red[512];
  float local = 0.f;
  for (int r = threadIdx.x; r < NROWS; r += 512) {
    const float* pm = &pmax[r * NCHUNKS];
    const float* ps = &psum[r * NCHUNKS];
    float M = -3.4e38f;
    for (int c = 0; c < NCHUNKS; ++c) M = fmaxf(M, pm[c]);
    float S = 0.f;
    for (int c = 0; c < NCHUNKS; ++c) S += ps[c] * __expf(pm[c] - M);
    local += -(tlogit[r] - M - logf(S));
  }
  red[threadIdx.x] = local;
  __syncthreads();
  for (int s = 256; s > 0; s >>= 1) {
    if (threadIdx.x < s) red[threadIdx.x] += red[threadIdx.x + s];
    __syncthreads();
  }
  if (threadIdx.x == 0) out[0] = red[0] / (float)NROWS;
}

// ---------------- launch ------------------------------------------------------
extern "C" void kernel_launch(void* const* d_in, const int* in_sizes, int n_in,
                              void* d_out, int out_size, void* d_ws, size_t ws_size,
                              hipStream_t stream) {
  (void)in_sizes; (void)n_in; (void)out_size; (void)ws_size;
  const int*   inputs  = (const int*)  d_in[0];
  const int*   targets = (const int*)  d_in[1];
  const float* Wih     = (const float*)d_in[2];
  const float* bih     = (const float*)d_in[3];
  const float* Whh     = (const float*)d_in[4];
  const float* bhh     = (const float*)d_in[5];
  const float* Wfc     = (const float*)d_in[6];
  const float* bfc     = (const float*)d_in[7];
  float* out = (float*)d_out;

  float* xproj = (float*)d_ws;                   // 4096*256
  float* hs    = xproj + NROWS * HIDDEN;         // 4096*256
  float* pmax  = hs    + NROWS * HIDDEN;         // 4096*125
  float* psum  = pmax  + NROWS * NCHUNKS;        // 4096*125
  float* tlog  = psum  + NROWS * NCHUNKS;        // 4096

  gather_kernel<<<NROWS, HIDDEN, 0, stream>>>(inputs, Wih, bih, bhh, xproj);

  const size_t rnn_lds = (size_t)(HIDDEN * LDA + 2 * BATCH * LDA) * sizeof(float);
  rnn_kernel<<<1, 512, rnn_lds, stream>>>(xproj, Whh, hs);

  const size_t fc_lds = (size_t)(FC_ROWS * LDA + 2 * 16 * LDA + FC_ROWS * LDL) * sizeof(float);
  dim3 fgrid(NROWS / FC_ROWS, NCHUNKS);
  fc_kernel<<<fgrid, 256, fc_lds, stream>>>(hs, Wfc, bfc, targets, pmax, psum, tlog);

  loss_kernel<<<1, 512, 0, stream>>>(pmax, psum, tlog, out);
}